// TiledPositionEncoder_42288247996732
// MI455X (gfx1250) — compile-verified
//
#include <hip/hip_runtime.h>

// ext-vector types (work with __builtin_nontemporal_store and WMMA builtins)
typedef float v2f __attribute__((ext_vector_type(2)));
typedef float v4f __attribute__((ext_vector_type(4)));
typedef float v8f __attribute__((ext_vector_type(8)));

#define STEPS  64
#define NPH    32            // phases per axis
#define DPTOT  96            // 3 axes * 32 phases
#define NTOK   (STEPS * STEPS * STEPS)   // 262144 grid tokens (cls handled separately)

// Each wave: one (16-token tile, axis d, 16-phase half) unit.
//   theta tile (16x16) = pos[16x1] x phases[1x16]  computed via V_WMMA_F32_16X16X4_F32
//   (rank-1 GEMM: only K=0 populated in A and B).
// D layout: VGPR v, lanes 0-15 -> token row M=v, lanes 16-31 -> M=v+8; N = lane&15 = phase.
// Each lane then emits 8 float4 {c,-s,s,c} blocks as b128 non-temporal stores.
__global__ __launch_bounds__(256) void rope_wmma_kernel(const float* __restrict__ phases,
                                                        float* __restrict__ out) {
    const int lane      = threadIdx.x & 31;
    const int waveInBlk = threadIdx.x >> 5;
    const int w         = blockIdx.x * 8 + waveInBlk;   // 0 .. 98303
    const int tileIdx   = w / 6;                        // 0 .. 16383 (16-token tile)
    const int rem       = w % 6;
    const int d         = rem >> 1;                     // axis 0..2
    const int ph        = rem & 1;                      // phase half 0..1
    const int tBase     = tileIdx * 16;                 // 0-based token (pre-cls)

    // ---- A matrix (16x4 f32): lanes 0-15 hold K=0 (pos of token M=lane); all else 0.
    const int   n     = tBase + (lane & 15);
    const int   ci    = n >> 12;          // x coord (i)
    const int   cj    = (n >> 6) & 63;    // y coord (j)
    const int   ck    = n & 63;           // z coord (k)
    const int   coord = (d == 0) ? ci : ((d == 1) ? cj : ck);
    const float pos   = (float)coord * (1.0f / 63.0f); // linspace(0,1,64)
    v2f a;
    a.x = (lane < 16) ? pos : 0.0f;  // lanes 0-15: K=0; lanes 16-31 would be K=2 -> 0
    a.y = 0.0f;                      // K=1 / K=3 -> 0

    // ---- B matrix (4x16 f32): row K=0 (lanes 0-15) = phases[d, ph*16 + lane]; rest 0.
    const float phv = phases[d * NPH + ph * 16 + (lane & 15)];
    v2f b;
    b.x = (lane < 16) ? phv : 0.0f;
    b.y = 0.0f;

    // ---- theta tile = A x B + 0  (straight-line code: EXEC all-1s as WMMA requires)
    v8f cacc = {};
    v8f theta = __builtin_amdgcn_wmma_f32_16x16x4_f32(
        /*neg_a=*/false, a, /*neg_b=*/false, b,
        /*c_mod=*/(short)0, cacc, /*reuse_a=*/false, /*reuse_b=*/false);

    // ---- emit rotation blocks: out[token][d*32 + p] = {cos, -sin, sin, cos}
    const int p    = ph * 16 + (lane & 15);
    const int mHi  = (lane >> 4) << 3;      // lanes 16-31 cover token rows M+8
    v4f* __restrict__ outv = (v4f*)out;
#pragma unroll
    for (int v = 0; v < 8; ++v) {
        const float th = theta[v];
        const float s  = __sinf(th);        // v_sin_f32 (theta in [0,1] rad)
        const float c  = __cosf(th);        // v_cos_f32
        const int token = tBase + v + mHi + 1;   // +1: cls occupies index 0
        v4f blk;
        blk.x = c; blk.y = -s; blk.z = s; blk.w = c;
        // 402 MB single-pass output > 192 MB L2: non-temporal b128 store
        __builtin_nontemporal_store(blk, &outv[(size_t)token * DPTOT + d * NPH + p]);
    }
}

// cls token (index 0): matrix_exp(0) = I for all 96 blocks
__global__ void rope_cls_kernel(float* __restrict__ out) {
    const int dp = threadIdx.x;
    if (dp < DPTOT) {
        v4f blk;
        blk.x = 1.0f; blk.y = 0.0f; blk.z = 0.0f; blk.w = 1.0f;
        __builtin_nontemporal_store(blk, &((v4f*)out)[dp]);
    }
}

extern "C" void kernel_launch(void* const* d_in, const int* in_sizes, int n_in,
                              void* d_out, int out_size, void* d_ws, size_t ws_size,
                              hipStream_t stream) {
    // d_in[0] = image_sizes (int32[1,3]) -- sizes match IMAGE_SIZE so scale==1, unused.
    const float* phases = (const float*)d_in[1];   // float32 [3,32]
    float* out = (float*)d_out;                    // float32 [1, 262145, 96, 2, 2]

    rope_cls_kernel<<<1, 128, 0, stream>>>(out);

    // 16384 token tiles * (3 axes * 2 phase-halves) = 98304 waves; 8 waves/block
    const int waves  = (NTOK / 16) * 6;   // 98304
    const int blocks = waves / 8;         // 12288
    rope_wmma_kernel<<<blocks, 256, 0, stream>>>(phases, out);
}